// Hierarical_CELoss1_82489141887107
// MI455X (gfx1250) — compile-verified
//
#include <hip/hip_runtime.h>
#include <hip/hip_bf16.h>
#include <stdint.h>

#define BATCH     32768
#define NCLS      1000
#define NF4       250        // float4s per row (1000/4)
#define CHUNK_F4  32         // one float4 per lane per chunk
#define NCHUNK    8          // ceil(250/32)
#define WAVES_PER_BLK 8

// ---------------- CDNA5 async global->LDS path (inline asm) ----------------
// ISA 15.18.3: GLOBAL_LOAD_ASYNC_TO_LDS_B128 (opcode 98), GV mode:
//   copies 16 bytes from the 64-bit global address to LDS at address VDST   (tracked by ASYNCcnt)
// Generic shared pointers carry the LDS logical address in their low 32 bits
// (ISA 10.2 aperture table), so truncation yields a valid VDST value.
__device__ __forceinline__ void async_copy_b128(const void* gptr, void* lptr) {
    uint32_t lds_off = (uint32_t)(uintptr_t)lptr;
    uint64_t gaddr   = (uint64_t)(uintptr_t)gptr;
    asm volatile("global_load_async_to_lds_b128 %0, %1, off"
                 :: "v"(lds_off), "v"(gaddr)
                 : "memory");
}

template <int N>
__device__ __forceinline__ void wait_async_le() {
    asm volatile("s_wait_asynccnt %0" :: "n"(N) : "memory");
}

// ---------------- kernel 0: init accumulators ------------------------------
__global__ void init_kernel(float* out, unsigned* maxbits) {
    out[0]     = 0.0f;
    maxbits[0] = 0u;     // +0.0f; distances are uniform[0,1) >= 0
}

// ---------------- kernel 1: max(distance_matrix) ---------------------------
__global__ __launch_bounds__(256) void maxdist_kernel(
    const float* __restrict__ dist, unsigned* __restrict__ maxbits, int n4) {
    int idx    = blockIdx.x * blockDim.x + threadIdx.x;
    int stride = gridDim.x * blockDim.x;
    float m = 0.0f;
    for (int i = idx; i < n4; i += stride) {
        float4 v = ((const float4*)dist)[i];
        m = fmaxf(m, fmaxf(fmaxf(v.x, v.y), fmaxf(v.z, v.w)));
    }
    #pragma unroll
    for (int off = 16; off > 0; off >>= 1)
        m = fmaxf(m, __shfl_xor(m, off, 32));

    __shared__ float wmax[WAVES_PER_BLK];
    int wv = threadIdx.x >> 5, lane = threadIdx.x & 31;
    if (lane == 0) wmax[wv] = m;
    __syncthreads();
    if (threadIdx.x == 0) {
        float bm = wmax[0];
        #pragma unroll
        for (int i = 1; i < WAVES_PER_BLK; ++i) bm = fmaxf(bm, wmax[i]);
        atomicMax(maxbits, __float_as_uint(bm));  // bit-order == float order (>=0)
    }
}

// ---------------- kernel 2: fused logsoftmax+argmax+loss -------------------
// One wave32 per row. Depth-2 async pipeline over 4 LDS buffers per wave.
__global__ __launch_bounds__(256) void loss_kernel(
    const float* __restrict__ y_pred, const int* __restrict__ y_true,
    const float* __restrict__ dist, float* __restrict__ out,
    const unsigned* __restrict__ maxbits) {

    __shared__ __align__(16) float lds[WAVES_PER_BLK][4][CHUNK_F4 * 4];  // 16 KB

    const int wv   = threadIdx.x >> 5;
    const int lane = threadIdx.x & 31;
    const int row  = blockIdx.x * WAVES_PER_BLK + wv;      // grid sized exactly
    const float* rowp = y_pred + (size_t)row * NCLS;
    float (*buf)[CHUNK_F4 * 4] = lds[wv];

    auto issue = [&](int chunk) {
        int f4 = chunk * CHUNK_F4 + lane;
        if (f4 > NF4 - 1) f4 = NF4 - 1;                    // clamp; masked in compute
        async_copy_b128(rowp + (size_t)f4 * 4, &buf[chunk & 3][lane * 4]);
    };

    // prime depth-2 pipeline
    issue(0);
    issue(1);

    float m = -INFINITY, s = 0.0f;       // online logsumexp state
    float bestv = -INFINITY;             // argmax state
    int   besti = 0x7fffffff;

    for (int c = 0; c < NCHUNK; ++c) {
        if (c == NCHUNK - 1) wait_async_le<0>(); else wait_async_le<1>();

        float4 v = *(const float4*)&buf[c & 3][lane * 4];  // ds_load_b128

        if (c + 2 < NCHUNK) issue(c + 2);  // writes buffer (c+2)&3 != c&3

        int f4o = c * CHUNK_F4 + lane;
        if (f4o < NF4) {
            int col0 = f4o * 4;
            float xs[4] = {v.x, v.y, v.z, v.w};
            #pragma unroll
            for (int j = 0; j < 4; ++j) {
                float x = xs[j];
                if (x > bestv) { bestv = x; besti = col0 + j; }  // first occurrence
                float mn = fmaxf(m, x);
                s = s * __expf(m - mn) + __expf(x - mn);
                m = mn;
            }
        }
    }

    // wave32 butterfly: (m,s) logsumexp merge + argmax with index tiebreak
    #pragma unroll
    for (int off = 16; off > 0; off >>= 1) {
        float om = __shfl_xor(m, off, 32);
        float os = __shfl_xor(s, off, 32);
        float mn = fmaxf(m, om);
        s = s * __expf(m - mn) + os * __expf(om - mn);
        m = mn;
        float obv = __shfl_xor(bestv, off, 32);
        int   obi = __shfl_xor(besti, off, 32);
        if (obv > bestv || (obv == bestv && obi < besti)) { bestv = obv; besti = obi; }
    }

    if (lane == 0) {
        int   t    = y_true[row];
        float xt   = rowp[t];                              // L2-resident re-gather
        float logp = xt - m - __logf(s);
        float maxd = __uint_as_float(maxbits[0]);
        float h    = dist[(size_t)besti * NCLS + t] / maxd;
        atomicAdd(out, -(logp * h) * (1.0f / (float)BATCH));
    }
}

// ---------------- launch ---------------------------------------------------
extern "C" void kernel_launch(void* const* d_in, const int* in_sizes, int n_in,
                              void* d_out, int out_size, void* d_ws, size_t ws_size,
                              hipStream_t stream) {
    const float* y_pred = (const float*)d_in[0];
    const int*   y_true = (const int*)d_in[1];    // jax default x64-off -> int32
    const float* dist   = (const float*)d_in[2];
    // d_in[3] (fix_layer) is dead code in the reference.
    float*    out     = (float*)d_out;
    unsigned* maxbits = (unsigned*)d_ws;

    init_kernel<<<1, 1, 0, stream>>>(out, maxbits);
    maxdist_kernel<<<256, 256, 0, stream>>>(dist, maxbits, (NCLS * NCLS) / 4);
    loss_kernel<<<BATCH / WAVES_PER_BLK, 256, 0, stream>>>(y_pred, y_true, dist, out, maxbits);
}